// costvolume_33363305955778
// MI455X (gfx1250) — compile-verified
//
#include <hip/hip_runtime.h>

// CDNA5 / gfx1250, wave32. f32 WMMA 16x16x4: A 16x4 (v2f), B 4x16 (v2f), C/D 16x16 (v8f).
typedef float v2f __attribute__((ext_vector_type(2)));
typedef float v4f __attribute__((ext_vector_type(4)));
typedef float v8f __attribute__((ext_vector_type(8)));

#define NB 4
#define NC 64
#define NH 128
#define NW 128
#define PLANE (NH * NW)     // 16384
#define IMG (NC * PLANE)    // per-batch stride of one (b,c,h,w) tensor

struct ViewPtrs { const float* p[16]; };

// ---- CDNA5 async global->LDS copies (ASYNCcnt) -----------------------------
__device__ __forceinline__ unsigned lds_off(const void* p) {
    // flat LDS-aperture address keeps the LDS byte offset in addr[31:0]
    return (unsigned)(unsigned long long)p;
}
__device__ __forceinline__ void async_ld_b128(unsigned lds, const float* gp) {
    asm volatile("global_load_async_to_lds_b128 %0, %1, off"
                 :: "v"(lds), "v"((unsigned long long)gp) : "memory");
}
__device__ __forceinline__ void async_ld_b32(unsigned lds, const float* gp) {
    asm volatile("global_load_async_to_lds_b32 %0, %1, off"
                 :: "v"(lds), "v"((unsigned long long)gp) : "memory");
}
__device__ __forceinline__ void wait_async0() {
    asm volatile("s_wait_asynccnt 0x0" ::: "memory");
}

// One wave (32 lanes) per block.
//  horizontal (vertical==0): blockIdx = (w-tile 0..7, h 0..127, b*4+k)
//  vertical   (vertical==1): blockIdx = (h-tile 0..7, w 0..127, b*4+k)
// Gram tiles G[p,j] = sum_c center[c,p]*view[c,j] via chained
// v_wmma_f32_16x16x4_f32 (K=64); diagonals m = sgn*(j-p) accumulate in LDS.
__global__ __launch_bounds__(32)
void costvol_wmma_kernel(const float* __restrict__ center, ViewPtrs vp,
                         float* __restrict__ out, int vertical)
{
    const int tid  = threadIdx.x;
    const int bk   = blockIdx.z;
    const int b    = bk >> 2;
    const int k    = bk & 3;
    const int d    = 4 * (k + 1);          // 4, 8, 12, 16
    const int Mk   = 2 * d + 1;            // 9, 17, 25, 33
    const int ntiles = (d <= 8) ? 2 : 3;   // window 16+2d pixels -> 2 or 3 j-tiles
    const int jw   = ntiles * 16;
    const int p0   = blockIdx.x * 16;      // tile start along the shift axis
    const int j0   = p0 - d;               // window start (can be <0; zero-padded)
    const int line = blockIdx.y;           // fixed coordinate on the other axis

    __shared__ float Alds[NC][16];         // center tile  [channel][pixel]
    __shared__ float Blds[2][NC][48];      // both view windows [view][channel][j]
    __shared__ float Olds[33 * 16];        // [m+d][pixel] accumulator

    const int cb      = b * IMG;
    const int lineoff = vertical ? line : line * NW;

    for (int idx = tid; idx < 33 * 16; idx += 32) Olds[idx] = 0.0f;

    // ---- Stage A + both view windows with async global->LDS copies ----------
    if (!vertical) {
        // contiguous rows: 16B chunks; d and image bounds are multiples of 4,
        // so each 4-float chunk is entirely in-image or entirely out.
        for (int cidx = tid; cidx < NC * 4; cidx += 32) {
            int c = cidx >> 2, qc = cidx & 3;
            async_ld_b128(lds_off(&Alds[c][qc * 4]),
                          center + cb + c * PLANE + lineoff + p0 + qc * 4);
        }
        const int cpr = jw >> 2;           // b128 chunks per row (8 or 12)
        for (int vi = 0; vi < 2; ++vi) {
            const float* view = vp.p[k * 4 + (vi ? 2 : 0)];
            for (int cidx = tid; cidx < NC * cpr; cidx += 32) {
                int c = cidx / cpr, qc = cidx - c * cpr;
                int j = j0 + qc * 4;
                if (j >= 0 && j < NW)
                    async_ld_b128(lds_off(&Blds[vi][c][qc * 4]),
                                  view + cb + c * PLANE + lineoff + j);
                else
                    *(v4f*)&Blds[vi][c][qc * 4] = (v4f){0.f, 0.f, 0.f, 0.f};
            }
        }
    } else {
        // column access (stride NW): per-element async b32
        for (int idx = tid; idx < NC * 16; idx += 32) {
            int c = idx >> 4, p = idx & 15;
            async_ld_b32(lds_off(&Alds[c][p]),
                         center + cb + c * PLANE + (p0 + p) * NW + line);
        }
        for (int vi = 0; vi < 2; ++vi) {
            const float* view = vp.p[k * 4 + (vi ? 1 : 3)];
            for (int idx = tid; idx < NC * jw; idx += 32) {
                int c = idx / jw, q = idx - c * jw;
                int j = j0 + q;
                if (j >= 0 && j < NH)
                    async_ld_b32(lds_off(&Blds[vi][c][q]),
                                 view + cb + c * PLANE + j * NW + line);
                else
                    Blds[vi][c][q] = 0.0f;
            }
        }
    }
    wait_async0();          // all tensor staging complete (ASYNCcnt == 0)
    __syncthreads();

    // ---- WMMA Gram tiles + diagonal scatter ---------------------------------
    const int half = tid >> 4;   // lane 0-15 vs 16-31 (K / M split per ISA layout)
    const int nn   = tid & 15;

    for (int vi = 0; vi < 2; ++vi) {
        // horiz: view i=0 (m=+s), i=2 (m=-s); vert: i=3 (m=+s), i=1 (m=-s)
        const int sgn = vi ? -1 : 1;
        for (int t = 0; t < ntiles; ++t) {
            v8f acc = {0.f, 0.f, 0.f, 0.f, 0.f, 0.f, 0.f, 0.f};
            #pragma unroll
            for (int kk = 0; kk < 16; ++kk) {          // K = 64 in chunks of 4
                const int c0 = kk * 4 + 2 * half;      // lanes 0-15: K=0,1; 16-31: K=2,3
                v2f a, bm;
                a.x  = Alds[c0][nn];                   // A[M=nn][K=c0]
                a.y  = Alds[c0 + 1][nn];               // A[M=nn][K=c0+1]
                bm.x = Blds[vi][c0][t * 16 + nn];      // B[K=c0][N=nn]
                bm.y = Blds[vi][c0 + 1][t * 16 + nn];  // B[K=c0+1][N=nn]
                acc = __builtin_amdgcn_wmma_f32_16x16x4_f32(
                        false, a, false, bm, (short)0, acc, false, false);
            }
            // acc[r] = G[M = r + 8*half][N = nn]; scatter diagonal m = sgn*(j-p)
            const int j = j0 + t * 16 + nn;
            #pragma unroll
            for (int r = 0; r < 8; ++r) {
                int M = r + 8 * half;                  // pixel within tile
                int m = sgn * (j - (p0 + M));
                if (m >= -d && m <= d)
                    Olds[(m + d) * 16 + M] += acc[r];  // distinct addr per lane
            }
        }
    }
    __syncthreads();

    // ---- Write / accumulate output ------------------------------------------
    int obase = 0;
    {
        const int mt[4] = {9, 17, 25, 33};
        for (int i = 0; i < k; ++i) obase += NB * mt[i] * PLANE;
    }
    for (int idx = tid; idx < Mk * 16; idx += 32) {
        int mi = idx >> 4, p = idx & 15;
        int hh = vertical ? (p0 + p) : line;
        int ww = vertical ? line : (p0 + p);
        int oidx = obase + ((b * Mk + mi) * NH + hh) * NW + ww;
        float val = Olds[idx] * (1.0f / 256.0f);       // mean over 64 ch, /4 views
        if (vertical) out[oidx] += val;                // pass 2 accumulates
        else          out[oidx]  = val;                // pass 1 initializes
    }
}

extern "C" void kernel_launch(void* const* d_in, const int* in_sizes, int n_in,
                              void* d_out, int out_size, void* d_ws, size_t ws_size,
                              hipStream_t stream) {
    (void)in_sizes; (void)n_in; (void)out_size; (void)d_ws; (void)ws_size;
    const float* center = (const float*)d_in[0];
    ViewPtrs vp;
    for (int i = 0; i < 16; ++i) vp.p[i] = (const float*)d_in[1 + i];
    float* out = (float*)d_out;

    dim3 grid(8, 128, 16);   // (tile, line, b*4+k)
    // Pass 1: horizontal shifts (views 0 & 2) — writes out.
    costvol_wmma_kernel<<<grid, 32, 0, stream>>>(center, vp, out, 0);
    // Pass 2: vertical shifts (views 1 & 3) — accumulates into out.
    costvol_wmma_kernel<<<grid, 32, 0, stream>>>(center, vp, out, 1);
}